// TopoEncoder_66803921322594
// MI455X (gfx1250) — compile-verified
//
#include <hip/hip_runtime.h>

typedef __attribute__((ext_vector_type(2))) float v2f;
typedef __attribute__((ext_vector_type(8))) float v8f;

// Hardware f32 atomic add (global_atomic_add_f32), no CAS fallback.
__device__ __forceinline__ void atom_add_f32(float* p, float v) {
  unsafeAtomicAdd(p, v);
}

// ---------------- utility kernels ----------------

__global__ void zero_f32(float* __restrict__ p, long long count) {
  long long t = (long long)blockIdx.x * blockDim.x + threadIdx.x;
  if (t < count) p[t] = 0.0f;
}

__global__ void degree_kernel(const int* __restrict__ dst, float* __restrict__ deg,
                              long long E) {
  long long t = (long long)blockIdx.x * blockDim.x + threadIdx.x;
  if (t < E) atom_add_f32(&deg[dst[t]], 1.0f);
}

__global__ void dinv_kernel(const float* __restrict__ deg, float* __restrict__ dinv, int n) {
  int t = blockIdx.x * blockDim.x + threadIdx.x;
  if (t < n) dinv[t] = rsqrtf(deg[t] + 1.0f);   // +1 accounts for the self-loop
}

// ---------------- WMMA fp32 GEMM: H[N,FOUT] = X[N,FIN] * W[FIN,FOUT] ----------------
// One wave (32 lanes) per 16-row M tile; wave holds FOUT/16 accumulators so X is
// streamed exactly once. Uses V_WMMA_F32_16X16X4_F32 (full fp32 precision).
//
// A 16x4 f32 layout: lane m = lane&15, kh = lane>>4; a[j] = X[m][k0 + 2*kh + j]
// B  4x16 f32 layout: lane n = lane&15;              b[j] = W[k0 + 2*kh + j][n]
// C 16x16 f32 layout: c[v] = C[v + 8*kh][n]

template <int FIN, int FOUT>
__global__ void gemm_wmma_f32(const float* __restrict__ X, const float* __restrict__ W,
                              float* __restrict__ H, int n) {
  constexpr int NT = FOUT / 16;
  const int wavesPerBlock = blockDim.x >> 5;
  const int wave = blockIdx.x * wavesPerBlock + (threadIdx.x >> 5);
  const int lane = threadIdx.x & 31;
  const int mtiles = n >> 4;
  if (wave >= mtiles) return;

  const int m   = (lane & 15);
  const int kh  = lane >> 4;
  const int col = (lane & 15);
  const float* xrow = X + (size_t)(wave * 16 + m) * FIN + 2 * kh;

  v8f acc[NT] = {};

  for (int k0 = 0; k0 < FIN; k0 += 4) {
    v2f a;
    a.x = xrow[k0 + 0];
    a.y = xrow[k0 + 1];
#pragma unroll
    for (int t = 0; t < NT; ++t) {
      v2f b;
      b.x = W[(size_t)(k0 + 2 * kh + 0) * FOUT + t * 16 + col];
      b.y = W[(size_t)(k0 + 2 * kh + 1) * FOUT + t * 16 + col];
      acc[t] = __builtin_amdgcn_wmma_f32_16x16x4_f32(
          /*neg_a=*/false, a, /*neg_b=*/false, b,
          /*c_mod=*/(short)0, acc[t], /*reuse_a=*/false, /*reuse_b=*/false);
    }
  }

#pragma unroll
  for (int t = 0; t < NT; ++t) {
#pragma unroll
    for (int v = 0; v < 8; ++v) {
      const int row = wave * 16 + v + 8 * kh;
      H[(size_t)row * FOUT + t * 16 + col] = acc[t][v];
    }
  }
}

// ---------------- edge scatter: out[dst] += h[src] * dinv[src]*dinv[dst] ----------------
// One thread per (edge, 4-feature chunk): float4 gather + 4 hardware f32 atomics.
// All h/agg buffers fit in the 192MB L2, so these atomics resolve at L2 rate.

template <int FOUT>
__global__ void scatter_edges(const float* __restrict__ H, const int* __restrict__ src,
                              const int* __restrict__ dst, const float* __restrict__ dinv,
                              float* __restrict__ out, long long E) {
  constexpr int C = FOUT / 4;
  long long t = (long long)blockIdx.x * blockDim.x + threadIdx.x;
  long long e = t / C;
  int c = (int)(t - e * C);
  if (e >= E) return;
  const int s = src[e];
  const int d = dst[e];
  const float w = dinv[s] * dinv[d];
  const float4 hv = *(const float4*)(H + (size_t)s * FOUT + 4 * c);
  float* o = out + (size_t)d * FOUT + 4 * c;
  atom_add_f32(o + 0, hv.x * w);
  atom_add_f32(o + 1, hv.y * w);
  atom_add_f32(o + 2, hv.z * w);
  atom_add_f32(o + 3, hv.w * w);
}

// ---------------- finalize: agg += h * dinv^2 (self-loop) + bias, optional ReLU ----------------

template <int FOUT, bool RELU>
__global__ void finalize_layer(float* __restrict__ agg, const float* __restrict__ H,
                               const float* __restrict__ dinv, const float* __restrict__ bias,
                               int n) {
  long long t = (long long)blockIdx.x * blockDim.x + threadIdx.x;
  if (t >= (long long)n * FOUT) return;
  const int i = (int)(t / FOUT);
  const int f = (int)(t - (long long)i * FOUT);
  const float di = dinv[i];
  float v = agg[t] + H[t] * di * di + bias[f];
  if (RELU) v = fmaxf(v, 0.0f);
  agg[t] = v;
}

// ---------------- launch ----------------

static inline long long cdiv_ll(long long a, long long b) { return (a + b - 1) / b; }

extern "C" void kernel_launch(void* const* d_in, const int* in_sizes, int n_in,
                              void* d_out, int out_size, void* d_ws, size_t ws_size,
                              hipStream_t stream) {
  const float* x  = (const float*)d_in[0];
  const int*   ei = (const int*)d_in[1];
  const float* W1 = (const float*)d_in[2];
  const float* b1 = (const float*)d_in[3];
  const float* W2 = (const float*)d_in[4];
  const float* b2 = (const float*)d_in[5];
  const float* W3 = (const float*)d_in[6];
  const float* b3 = (const float*)d_in[7];

  const int       N = in_sizes[0] / 128;     // 100000
  const long long E = in_sizes[1] / 2;       // 3200000
  const int* src = ei;
  const int* dst = ei + E;

  float* out = (float*)d_out;                // [N,16]

  // workspace layout
  float* deg  = (float*)d_ws;                // N
  float* dinv = deg + N;                     // N
  float* bufA = dinv + N;                    // N*64
  float* bufB = bufA + (size_t)N * 64;       // N*64
  float* bufC = bufB + (size_t)N * 64;       // N*32

  const int B = 256;
  const int mtiles = N / 16;
  const int gemmBlocks = (int)cdiv_ll(mtiles, B / 32);

  // --- normalization ---
  zero_f32<<<(int)cdiv_ll(N, B), B, 0, stream>>>(deg, N);
  degree_kernel<<<(int)cdiv_ll(E, B), B, 0, stream>>>(dst, deg, E);
  dinv_kernel<<<(int)cdiv_ll(N, B), B, 0, stream>>>(deg, dinv, N);

  // --- layer 1: 128 -> 64, ReLU ---
  gemm_wmma_f32<128, 64><<<gemmBlocks, B, 0, stream>>>(x, W1, bufB, N);
  zero_f32<<<(int)cdiv_ll((long long)N * 64, B), B, 0, stream>>>(bufA, (long long)N * 64);
  scatter_edges<64><<<(int)cdiv_ll(E * 16, B), B, 0, stream>>>(bufB, src, dst, dinv, bufA, E);
  finalize_layer<64, true><<<(int)cdiv_ll((long long)N * 64, B), B, 0, stream>>>(bufA, bufB, dinv, b1, N);

  // --- layer 2: 64 -> 32, ReLU ---
  gemm_wmma_f32<64, 32><<<gemmBlocks, B, 0, stream>>>(bufA, W2, bufC, N);
  zero_f32<<<(int)cdiv_ll((long long)N * 32, B), B, 0, stream>>>(bufB, (long long)N * 32);
  scatter_edges<32><<<(int)cdiv_ll(E * 8, B), B, 0, stream>>>(bufC, src, dst, dinv, bufB, E);
  finalize_layer<32, true><<<(int)cdiv_ll((long long)N * 32, B), B, 0, stream>>>(bufB, bufC, dinv, b2, N);

  // --- layer 3: 32 -> 16, no ReLU, aggregate directly into d_out ---
  gemm_wmma_f32<32, 16><<<gemmBlocks, B, 0, stream>>>(bufB, W3, bufC, N);
  zero_f32<<<(int)cdiv_ll((long long)N * 16, B), B, 0, stream>>>(out, (long long)N * 16);
  scatter_edges<16><<<(int)cdiv_ll(E * 4, B), B, 0, stream>>>(bufC, src, dst, dinv, out, E);
  finalize_layer<16, false><<<(int)cdiv_ll((long long)N * 16, B), B, 0, stream>>>(out, bufC, dinv, b3, N);
}